// feature_aggregation_layer_87187836109116
// MI455X (gfx1250) — compile-verified
//
#include <hip/hip_runtime.h>

#define BB   2
#define CC   128
#define NN   8192
#define KNN  20
#define KSEL 21   // knn_k + 1 (self included, dropped at the end)

typedef __attribute__((ext_vector_type(2))) float v2f;
typedef __attribute__((ext_vector_type(8))) float v8f;

// ---------------------------------------------------------------------------
// Phase 0: per-point squared norms  xx[b*N+n] = sum_c x[b,c,n]^2
// ---------------------------------------------------------------------------
__global__ __launch_bounds__(256) void fal_norms_kernel(const float* __restrict__ x,
                                                        float* __restrict__ xx) {
    int t = blockIdx.x * 256 + threadIdx.x;      // t in [0, B*N)
    int b = t / NN, n = t % NN;
    const float* p = x + (size_t)b * CC * NN + n;
    float s = 0.f;
    #pragma unroll 8
    for (int c = 0; c < CC; ++c) {
        float v = p[(size_t)c * NN];
        s += v * v;
    }
    xx[t] = s;
}

// ---------------------------------------------------------------------------
// Phase 1: fused fp32-WMMA distance GEMM + per-row top-21 selection.
// One workgroup = 32 rows of one batch; 8 waves sweep interleaved 16-wide
// column tiles. A-panel (32 rows x 128 K) is register-stationary: loaded once
// into 128 VGPRs, reused across all 64 column tiles per wave, so the hot loop
// is 2 B-loads + 2 WMMAs per k-step. score[m] = xx[m] - 2*<x_n,x_m>.
// ---------------------------------------------------------------------------
__global__ __launch_bounds__(256) void fal_knn_topk_kernel(const float* __restrict__ x,
                                                           const float* __restrict__ xx,
                                                           int* __restrict__ knn_idx) {
    const int b    = blockIdx.y;
    const int row0 = blockIdx.x * 32;
    const int wave = threadIdx.x >> 5;   // 0..7
    const int lane = threadIdx.x & 31;
    const int half = lane >> 4;          // 0: lanes 0-15, 1: lanes 16-31
    const int l16  = lane & 15;

    const float* xb = x + (size_t)b * CC * NN;

    __shared__ float s_scores[8][32][17];          // padded: conflict-free row reads
    __shared__ float s_md[32][8][KSEL];            // per-row per-wave sorted dists
    __shared__ int   s_mi[32][8][KSEL];            // per-row per-wave sorted cols

    // per-lane sorted top-KSEL list; lane owns global row (row0 + lane)
    float ld[KSEL];
    int   li[KSEL];
    #pragma unroll
    for (int i = 0; i < KSEL; ++i) { ld[i] = __builtin_inff(); li[i] = 0x7fffffff; }

    // ---- A-panel, register stationary (fp32 A 16x4 layout: lane -> M = l16,
    //      K pair base = half*2; VGPR0 = K0/K2, VGPR1 = K1/K3) ----
    v2f a0r[CC / 4];   // rows row0 +  0..15
    v2f a1r[CC / 4];   // rows row0 + 16..31
    {
        const float* pa0 = xb + (size_t)(half * 2) * NN + (row0 + l16);
        const float* pa1 = pa0 + 16;
        #pragma unroll
        for (int ks = 0; ks < CC / 4; ++ks) {
            a0r[ks].x = pa0[0];  a0r[ks].y = pa0[NN];
            a1r[ks].x = pa1[0];  a1r[ks].y = pa1[NN];
            pa0 += (size_t)4 * NN;  pa1 += (size_t)4 * NN;
        }
    }

    for (int tile = wave; tile < NN / 16; tile += 8) {
        const int col0 = tile * 16;
        const float xxc = xx[b * NN + col0 + l16];   // issued early, used in epilogue
        v8f c0 = {}; v8f c1 = {};
        const float* pb = xb + (size_t)(half * 2) * NN + (col0 + l16);

        #pragma unroll        // full unroll: a*r[] indices must be constants
        for (int ks = 0; ks < CC / 4; ++ks) {
            v2f bf;
            bf.x = pb[0];  bf.y = pb[NN];
            pb += (size_t)4 * NN;
            // D = A x B + C, fp32 tensor op (chained accumulate over K=128)
            c0 = __builtin_amdgcn_wmma_f32_16x16x4_f32(false, a0r[ks], false, bf,
                                                       (short)0, c0, false, false);
            c1 = __builtin_amdgcn_wmma_f32_16x16x4_f32(false, a1r[ks], false, bf,
                                                       (short)0, c1, false, false);
        }

        // Epilogue: score = xx[col] - 2*inner, transpose via LDS (this wave only)
        #pragma unroll
        for (int v = 0; v < 8; ++v) {
            const int r = v + half * 8;              // D layout: lanes16-31 -> M=v+8
            s_scores[wave][r][l16]      = __builtin_fmaf(-2.0f, c0[v], xxc);
            s_scores[wave][16 + r][l16] = __builtin_fmaf(-2.0f, c1[v], xxc);
        }
        __builtin_amdgcn_wave_barrier();             // LDS in-order within wave

        // Scan: lane owns row `lane`; 16 candidates, ascending col (tie -> low idx)
        float worst = ld[KSEL - 1];
        #pragma unroll 4
        for (int j = 0; j < 16; ++j) {
            const float s = s_scores[wave][lane][j];
            if (s < worst) {
                float cd = s; int ci = col0 + j;
                #pragma unroll
                for (int i = 0; i < KSEL; ++i) {     // branchless bubble insertion
                    const bool sw = cd < ld[i];
                    const float td = ld[i]; const int ti = li[i];
                    ld[i] = sw ? cd : td;  li[i] = sw ? ci : ti;
                    cd    = sw ? td : cd;  ci    = sw ? ti : ci;
                }
                worst = ld[KSEL - 1];
            }
        }
        __builtin_amdgcn_wave_barrier();             // WAR before next tile's writes
    }

    // dump per-wave sorted lists for the cross-wave merge
    #pragma unroll
    for (int i = 0; i < KSEL; ++i) {
        s_md[lane][wave][i] = ld[i];
        s_mi[lane][wave][i] = li[i];
    }
    __syncthreads();

    // 8-way merge of sorted lists, lexicographic (dist, idx); drop rank 0 (self)
    if (threadIdx.x < 32) {
        const int r = threadIdx.x;
        int h[8] = {0, 0, 0, 0, 0, 0, 0, 0};
        int* outp = knn_idx + ((size_t)b * NN + (row0 + r)) * KNN;
        for (int o = 0; o < KSEL; ++o) {
            float bd = __builtin_inff(); int bi = 0x7fffffff; int bw = 0;
            #pragma unroll
            for (int w = 0; w < 8; ++w) {
                const float d = s_md[r][w][h[w]];
                const int   i = s_mi[r][w][h[w]];
                const bool better = (d < bd) || (d == bd && i < bi);
                bd = better ? d : bd; bi = better ? i : bi; bw = better ? w : bw;
            }
            #pragma unroll
            for (int w = 0; w < 8; ++w) h[w] += (w == bw) ? 1 : 0;
            if (o > 0) outp[o - 1] = bi;
        }
    }
}

// ---------------------------------------------------------------------------
// Phase 2: out[b,c,n] = max_j x[b,c,knn[b,n,j]]   (self term cancels exactly)
// ---------------------------------------------------------------------------
__global__ __launch_bounds__(256) void fal_gather_max_kernel(const float* __restrict__ x,
                                                             const int* __restrict__ knn_idx,
                                                             float* __restrict__ out) {
    const int b = blockIdx.y;
    const int n = blockIdx.x * 256 + threadIdx.x;
    int idx[KNN];
    const int* ip = knn_idx + ((size_t)b * NN + n) * KNN;
    #pragma unroll
    for (int j = 0; j < KNN; ++j) idx[j] = ip[j];

    const float* xb = x   + (size_t)b * CC * NN;
    float*       ob = out + (size_t)b * CC * NN;
    for (int c = 0; c < CC; ++c) {                   // all threads share one 32KB row
        const float* row = xb + (size_t)c * NN;
        float m = -__builtin_inff();
        #pragma unroll
        for (int j = 0; j < KNN; ++j) m = fmaxf(m, row[idx[j]]);
        ob[(size_t)c * NN + n] = m;
    }
}

// ---------------------------------------------------------------------------
extern "C" void kernel_launch(void* const* d_in, const int* in_sizes, int n_in,
                              void* d_out, int out_size, void* d_ws, size_t ws_size,
                              hipStream_t stream) {
    const float* x = (const float*)d_in[0];          // vertex_feat (B,C,N) f32

    float* xx      = (float*)d_ws;                                   // B*N f32
    int*   knn_idx = (int*)((char*)d_ws + (size_t)BB * NN * sizeof(float)); // B*N*20 i32
    float* out     = (float*)d_out;

    fal_norms_kernel<<<dim3((BB * NN) / 256), dim3(256), 0, stream>>>(x, xx);
    fal_knn_topk_kernel<<<dim3(NN / 32, BB), dim3(256), 0, stream>>>(x, xx, knn_idx);
    fal_gather_max_kernel<<<dim3(NN / 256, BB), dim3(256), 0, stream>>>(x, knn_idx, out);
}